// FLAOperator_45492293599757
// MI455X (gfx1250) — compile-verified
//
#include <hip/hip_runtime.h>
#include <math.h>

// Gated linear attention (chunked) for B=4,H=16,N=2048,D=64 on gfx1250.
// One block per (head, value-half). 4 waves/block, chunk=64, f32 WMMA.
// q/k/v tiles staged into LDS by the Tensor Data Mover (async, double-buffered,
// with D# padding used to produce bank-conflict-free LDS row strides).

typedef __attribute__((ext_vector_type(2))) float v2f;
typedef __attribute__((ext_vector_type(8))) float v8f;
typedef __attribute__((ext_vector_type(4))) unsigned int u32x4;
typedef __attribute__((ext_vector_type(8))) int i32x8;
typedef __attribute__((ext_vector_type(4))) int i32x4;

#define NSEQ    2048
#define DK      64
#define CHUNK   64
#define NCHUNK  (NSEQ / CHUNK)
#define EHALF   32
#define QKS     76   // 64 DW row + 12 DW pad (TDM pad_interval=5, pad_amount=11)
#define VSS     44   // 32 DW row + 12 DW pad (TDM pad_interval=4, pad_amount=11)
#define SCALE   0.125f  // 1/sqrt(64)

__device__ __forceinline__ v8f wmma4(v2f a, v2f b, v8f c) {
  // D = A(16x4,f32) * B(4x16,f32) + C(16x16,f32)
  return __builtin_amdgcn_wmma_f32_16x16x4_f32(false, a, false, b, (short)0, c,
                                               false, false);
}

// Issue one TDM 2D tile load: rows x tile_d0 f32 elements, row stride
// `stride_elems` in memory, destination LDS rows padded per pad codes.
__device__ __forceinline__ void tdm_load_2d(const float* gptr, const float* lptr,
                                            unsigned tensor_d0, unsigned tile_d0,
                                            unsigned rows, unsigned stride_elems,
                                            unsigned pad_interval_code,
                                            unsigned pad_amount_code) {
  unsigned long long ga = (unsigned long long)(uintptr_t)gptr;
  unsigned int lds = (unsigned int)(uintptr_t)lptr;  // LDS offset in low bits
  u32x4 g0;
  g0[0] = 1u;                                        // count=1 (valid), user mode
  g0[1] = lds;                                       // lds_addr (bytes)
  g0[2] = (unsigned int)(ga & 0xFFFFFFFFu);          // global_addr[31:0]
  g0[3] = (unsigned int)((ga >> 32) & 0x01FFFFFFu)   // global_addr[56:32]
          | (2u << 30);                              // type=2 ("image")
  i32x8 g1;
  g1[0] = (int)((2u << 16)                           // data_size=4B
                | (1u << 20)                         // pad_enable
                | (pad_interval_code << 22) | (pad_amount_code << 25));
  g1[1] = (int)((tensor_d0 & 0xFFFFu) << 16);        // tensor_dim0[15:0]
  g1[2] = (int)(((tensor_d0 >> 16) & 0xFFFFu) | ((rows & 0xFFFFu) << 16));
  g1[3] = (int)(((rows >> 16) & 0xFFFFu) | (tile_d0 << 16));   // tile_dim0
  g1[4] = (int)(rows & 0xFFFFu);                     // tile_dim1=rows, tile_dim2=0
  g1[5] = (int)stride_elems;                         // tensor_dim0_stride[31:0]
  g1[6] = 0;
  g1[7] = 0;
  i32x4 z4 = {0, 0, 0, 0};                           // 2D tensor: groups 2/3 unused
  i32x8 z8 = {0, 0, 0, 0, 0, 0, 0, 0};
  __builtin_amdgcn_tensor_load_to_lds(g0, g1, z4, z4, z8, 0);
}

__global__ __launch_bounds__(128) void gla_chunk_kernel(
    const float* __restrict__ q, const float* __restrict__ k,
    const float* __restrict__ v, const float* __restrict__ g,
    float* __restrict__ out) {
  __shared__ float q_s[2][64 * QKS];
  __shared__ float k_s[2][64 * QKS];
  __shared__ float v_s[2][64 * VSS];
  __shared__ float A_s[64 * QKS];
  __shared__ float S_s[64 * VSS];   // running state S (DK x EHALF)
  __shared__ float Lam_s[64];       // inclusive cumsum of log(lambda) in chunk
  __shared__ float wfac_s[64];      // 1 - lambda
  __shared__ float coef_s[64];      // exp(Lam63 - Lam_t) * (1-lambda_t)
  __shared__ float eLam_s[64];      // exp(Lam_t)

  const int tid  = threadIdx.x;
  const int wv   = tid >> 5;        // wave 0..3 -> time rows [16*wv, 16*wv+16)
  const int lane = tid & 31;
  const int hs   = lane >> 4;       // half-wave select (K pair / M+8 select)
  const int lid  = lane & 15;

  const int bh = blockIdx.x >> 1;
  const int e0 = (blockIdx.x & 1) * EHALF;
  const size_t base  = (size_t)bh * NSEQ * DK;
  const size_t gbase = (size_t)bh * NSEQ;

  // Zero the state (includes padding; harmless).
  for (int i = tid; i < 64 * VSS; i += 128) S_s[i] = 0.0f;

  // Prologue: async-stage chunk 0 (wave 0 issues; TENSORcnt tracks).
  if (wv == 0) {
    tdm_load_2d(q + base,      q_s[0], 64, 64, 64, 64, 5, 11);
    tdm_load_2d(k + base,      k_s[0], 64, 64, 64, 64, 5, 11);
    tdm_load_2d(v + base + e0, v_s[0], 32, 32, 64, 64, 4, 11);
  }
  // Gate pipeline: register-prefetch g for chunk 0.
  float gv = (tid < 64) ? g[gbase + tid] : 0.0f;

  for (int c = 0; c < NCHUNK; ++c) {
    const int t0  = c * CHUNK;
    const int cur = c & 1;

    // ---- Issue next chunk's TDM loads, then wait for current chunk ----
    if (wv == 0) {
      if (c + 1 < NCHUNK) {
        const size_t nt0 = (size_t)(t0 + CHUNK) * DK;
        tdm_load_2d(q + base + nt0,      q_s[cur ^ 1], 64, 64, 64, 64, 5, 11);
        tdm_load_2d(k + base + nt0,      k_s[cur ^ 1], 64, 64, 64, 64, 5, 11);
        tdm_load_2d(v + base + nt0 + e0, v_s[cur ^ 1], 32, 32, 64, 64, 4, 11);
        __builtin_amdgcn_s_wait_tensorcnt(3);  // oldest 3 (= current chunk) done
      } else {
        __builtin_amdgcn_s_wait_tensorcnt(0);
      }
    }
    // ---- Gate prep for this chunk ----
    if (tid < 64) {
      float lam   = 1.0f / (1.0f + __expf(-gv));
      wfac_s[tid] = 1.0f - lam;
      Lam_s[tid]  = __logf(lam);
    }
    if (c + 1 < NCHUNK && tid < 64)  // prefetch next chunk's gate
      gv = g[gbase + t0 + CHUNK + tid];
    __syncthreads();                 // TDM data + gate values visible to all
    if (tid == 0) {                  // serial inclusive prefix (64 steps, cheap)
      float acc = 0.0f;
      for (int t = 0; t < 64; ++t) { acc += Lam_s[t]; Lam_s[t] = acc; }
    }
    __syncthreads();
    if (tid < 64) {
      coef_s[tid] = __expf(Lam_s[63] - Lam_s[tid]) * wfac_s[tid];
      eLam_s[tid] = __expf(Lam_s[tid]);
    }
    __syncthreads();

    const float* ql = q_s[cur];
    const float* kl = k_s[cur];
    const float* vl = v_s[cur];

    // ---- Phase A: A = scale*(Q K^T) (16x64 strip/wave), decay-mask, to LDS ----
    {
      const float* qrow = &ql[(16 * wv + lid) * QKS];
      for (int nt = 0; nt < 4; ++nt) {
        const float* krow = &kl[(16 * nt + lid) * QKS];
        v8f acc = {};
#pragma unroll
        for (int k0 = 0; k0 < 64; k0 += 4) {
          v2f a = *(const v2f*)&qrow[k0 + 2 * hs];
          v2f b = *(const v2f*)&krow[k0 + 2 * hs];
          acc = wmma4(a, b, acc);
        }
        const int i     = 16 * nt + lid;
        const float li  = Lam_s[i];
        const float wfS = wfac_s[i] * SCALE;
#pragma unroll
        for (int r = 0; r < 8; ++r) {
          int t = 16 * wv + r + 8 * hs;
          // Branchless causal decay mask: exp on clamped arg (no inf*0), then select.
          float e = __expf(fminf(Lam_s[t] - li, 0.0f)) * wfS;
          float m = (i <= t) ? e : 0.0f;
          A_s[t * QKS + i] = acc[r] * m;
        }
      }
    }
    __syncthreads();

    // ---- Phase B: O = (A V) + diag(exp(Lam)) (scale*Q) S_in ; store global ----
    {
      const float* qrow = &ql[(16 * wv + lid) * QKS];
      const float* arow = &A_s[(16 * wv + lid) * QKS];
      const float  pt   = eLam_s[16 * wv + lid] * SCALE;
      for (int et = 0; et < 2; ++et) {
        const int col = 16 * et + lid;
        v8f acc = {};
#pragma unroll
        for (int k0 = 0; k0 < 64; k0 += 4) {  // intra-chunk: A * V
          v2f a = *(const v2f*)&arow[k0 + 2 * hs];
          v2f b;
          b.x = vl[(k0 + 2 * hs) * VSS + col];
          b.y = vl[(k0 + 2 * hs + 1) * VSS + col];
          acc = wmma4(a, b, acc);
        }
#pragma unroll
        for (int k0 = 0; k0 < 64; k0 += 4) {  // inter-chunk: (P.Q) * S_in
          v2f a = *(const v2f*)&qrow[k0 + 2 * hs];
          a.x *= pt; a.y *= pt;
          v2f b;
          b.x = S_s[(k0 + 2 * hs) * VSS + col];
          b.y = S_s[(k0 + 2 * hs + 1) * VSS + col];
          acc = wmma4(a, b, acc);
        }
#pragma unroll
        for (int r = 0; r < 8; ++r) {
          int t = 16 * wv + r + 8 * hs;
          out[base + (size_t)(t0 + t) * DK + e0 + 16 * et + lid] = acc[r];
        }
      }
    }
    __syncthreads();

    // ---- Phase C: S_out = exp(Lam63) * S_in + Khat^T V ----
    {
      const float p63 = eLam_s[63];
      for (int et = 0; et < 2; ++et) {
        const int col = 16 * et + lid;
        v8f acc;
#pragma unroll
        for (int r = 0; r < 8; ++r)
          acc[r] = p63 * S_s[(16 * wv + r + 8 * hs) * VSS + col];
#pragma unroll
        for (int k0 = 0; k0 < 64; k0 += 4) {
          int tA = k0 + 2 * hs;
          v2f a, b;
          a.x = coef_s[tA]     * kl[tA * QKS       + 16 * wv + lid];
          a.y = coef_s[tA + 1] * kl[(tA + 1) * QKS + 16 * wv + lid];
          b.x = vl[tA * VSS + col];
          b.y = vl[(tA + 1) * VSS + col];
          acc = wmma4(a, b, acc);
        }
#pragma unroll
        for (int r = 0; r < 8; ++r)
          S_s[(16 * wv + r + 8 * hs) * VSS + col] = acc[r];
      }
    }
    __syncthreads();  // protect stage buffers / S_s before next chunk
  }
}

extern "C" void kernel_launch(void* const* d_in, const int* in_sizes, int n_in,
                              void* d_out, int out_size, void* d_ws, size_t ws_size,
                              hipStream_t stream) {
  const float* q = (const float*)d_in[0];
  const float* k = (const float*)d_in[1];
  const float* v = (const float*)d_in[2];
  const float* g = (const float*)d_in[3];
  float* out = (float*)d_out;
  (void)in_sizes; (void)n_in; (void)out_size; (void)d_ws; (void)ws_size;
  dim3 grid(4 * 16 * 2);  // (B*H) heads x 2 value-halves = 128 blocks
  dim3 block(128);        // 4 waves
  gla_chunk_kernel<<<grid, block, 0, stream>>>(q, k, v, g, out);
}